// LATTE_43361989820788
// MI455X (gfx1250) — compile-verified
//
#include <hip/hip_runtime.h>
#include <hip/hip_bf16.h>

// ---- problem constants (match reference) ----
#define NN 50000
#define EE 800000
#define MM 4
#define HH 4
#define CC 32
#define INF_ 256
#define DD 128   // HH*CC
#define NEG_SLOPE 0.2f
#define LDS_STRIDE 260   // 256 + 4 pad floats per row (TDM pad: 16B per 1KB)

typedef __attribute__((ext_vector_type(2))) float v2f;
typedef __attribute__((ext_vector_type(8))) float v8f;
typedef __attribute__((ext_vector_type(4))) unsigned int v4u;
typedef __attribute__((ext_vector_type(4))) int v4i;
typedef __attribute__((ext_vector_type(8))) int v8i;

__device__ __forceinline__ float lrelu(float v) { return v > 0.f ? v : NEG_SLOPE * v; }

// ---------------------------------------------------------------------------
// 0) init: maxb = -FLT_MAX, sumb = 0, rel = 0 (must re-init every call)
// ---------------------------------------------------------------------------
__global__ void latte_init(float* __restrict__ maxb, float* __restrict__ sumb,
                           float* __restrict__ rel) {
  const long SC = (long)MM * NN * HH;          // 800,000
  const long RC = (long)MM * NN * DD;          // 25,600,000
  long total = 2 * SC + RC;
  for (long i = (long)blockIdx.x * blockDim.x + threadIdx.x; i < total;
       i += (long)gridDim.x * blockDim.x) {
    if (i < SC)            maxb[i] = -3.402823466e38f;
    else if (i < 2 * SC)   sumb[i - SC] = 0.f;
    else                   rel[i - 2 * SC] = 0.f;
  }
}

// ---------------------------------------------------------------------------
// 1) projection: h = relu(feats @ W + b), f32 WMMA 16x16x4.
//    One wave per 16-row panel computing all 128 output columns (8 tiles).
//    A panel (16x256) staged into LDS by ONE Tensor-Data-Mover descriptor
//    (pad_enable inserts 16B per 1KB row -> LDS row stride 260 floats,
//    avoiding the 16-way bank conflict of stride-256 access). B from L2.
//    A frag (16x4):  lanes0-15 M=0..15 {v0:K0,v1:K1}; lanes16-31 {v0:K2,v1:K3}
//    B frag (4x16):  lanes0-15 N=0..15 {v0:K0,v1:K1}; lanes16-31 {v0:K2,v1:K3}
//    C/D (16x16):    vj: lanes0-15 M=j, lanes16-31 M=8+j, N=lane&15
// ---------------------------------------------------------------------------
__global__ void latte_proj_wmma(const float* __restrict__ feats,
                                const float* __restrict__ W,
                                const float* __restrict__ bias,
                                float* __restrict__ h) {
  __shared__ float As[16 * LDS_STRIDE];        // 16,640 B per (1-wave) block
  const int tile_m = blockIdx.x;               // 0..3124
  const int lane   = threadIdx.x;              // wave32
  const int lo = lane & 15;
  const int hi = lane >> 4;
  const float* gA = feats + (long)tile_m * 16 * INF_;

#if defined(__gfx1250__) && __has_builtin(__builtin_amdgcn_tensor_load_to_lds) && __has_builtin(__builtin_amdgcn_s_wait_tensorcnt)
  {
    const unsigned long long ga = (unsigned long long)(uintptr_t)gA;
    // D# group0: count=1 valid descriptor; lds_addr; 57-bit global addr; type=2
    v4u g0 = { 1u,
               (unsigned int)(uintptr_t)As,
               (unsigned int)ga,
               ((unsigned int)(ga >> 32) & 0x01FFFFFFu) | (2u << 30) };
    // D# group1: data_size=4B(2), pad_enable=1, pad_interval=7(256dw),
    //            pad_amount=3(4dw); tensor_dim0=256; tensor_dim1=16;
    //            tile_dim0=256; tile_dim1=16; tensor_dim0_stride=256
    v8i g1 = { (int)((2u << 16) | (1u << 20) | (7u << 22) | (3u << 25)),
               (int)(256u << 16),
               (int)(16u << 16),
               (int)(256u << 16),
               16,
               256,
               0, 0 };
    v4i z4 = {0, 0, 0, 0};
    v8i z8 = {0, 0, 0, 0, 0, 0, 0, 0};
    // 6-arg toolchain form: (g0, g1, g2, g3, g4, cpol)
    __builtin_amdgcn_tensor_load_to_lds(g0, g1, z4, z4, z8, 0);
    __builtin_amdgcn_s_wait_tensorcnt(0);
  }
#else
  for (int i = lane * 4; i < 16 * INF_; i += 32 * 4) {
    const int r = i / INF_, c = i % INF_;
    *(float4*)&As[r * LDS_STRIDE + c] = *(const float4*)&gA[i];
  }
#endif

  v8f acc[8];
#pragma unroll
  for (int t = 0; t < 8; ++t) acc[t] = (v8f){};

  for (int k0 = 0; k0 < INF_; k0 += 4) {
    const int k = k0 + 2 * hi;
    v2f a;
    a.x = As[lo * LDS_STRIDE + k];
    a.y = As[lo * LDS_STRIDE + k + 1];
#pragma unroll
    for (int t = 0; t < 8; ++t) {
      const int col = t * 16 + lo;
      v2f b;
      b.x = W[(long)k * DD + col];
      b.y = W[(long)(k + 1) * DD + col];
      acc[t] = __builtin_amdgcn_wmma_f32_16x16x4_f32(false, a, false, b,
                                                     (short)0, acc[t],
                                                     false, false);
    }
  }

#pragma unroll
  for (int t = 0; t < 8; ++t) {
    const int col = t * 16 + lo;
    const float bv = bias[col];
#pragma unroll
    for (int j = 0; j < 8; ++j) {
      const int r = tile_m * 16 + j + 8 * hi;
      float v = acc[t][j] + bv;
      h[(long)r * DD + col] = v > 0.f ? v : 0.f;
    }
  }
}

// ---------------------------------------------------------------------------
// 2) per-node attention scores: al[m,n,h] = <h[n,h,:], attn[m,h,:C]>
//                               ar[m,n,h] = <h[n,h,:], attn[m,h,C:]>
// ---------------------------------------------------------------------------
__global__ void latte_scores(const float* __restrict__ h,
                             const float* __restrict__ attn,
                             float* __restrict__ al, float* __restrict__ ar) {
  const int wave = threadIdx.x >> 5;
  const int lane = threadIdx.x & 31;
  const int n = blockIdx.x * 4 + wave;
  if (n >= NN) return;
  float hv[HH];
#pragma unroll
  for (int hd = 0; hd < HH; ++hd) hv[hd] = h[(long)n * DD + hd * CC + lane];
#pragma unroll
  for (int m = 0; m < MM; ++m) {
#pragma unroll
    for (int hd = 0; hd < HH; ++hd) {
      float l = hv[hd] * attn[(m * HH + hd) * (2 * CC) + lane];
      float r = hv[hd] * attn[(m * HH + hd) * (2 * CC) + CC + lane];
#pragma unroll
      for (int off = 16; off > 0; off >>= 1) {
        l += __shfl_xor(l, off, 32);
        r += __shfl_xor(r, off, 32);
      }
      if (lane == 0) {
        al[((long)m * NN + n) * HH + hd] = l;
        ar[((long)m * NN + n) * HH + hd] = r;
      }
    }
  }
}

// ---------------------------------------------------------------------------
// 3) edge pass 1: segment max of leaky_relu(al[dst] + ar[src]) per (m,dst,h)
// ---------------------------------------------------------------------------
__global__ void latte_edge_max(const int* __restrict__ ei,
                               const float* __restrict__ al,
                               const float* __restrict__ ar,
                               float* __restrict__ maxb) {
  long idx = (long)blockIdx.x * blockDim.x + threadIdx.x;
  if (idx >= (long)MM * EE) return;
  const int m = (int)(idx / EE);
  const int e = (int)(idx % EE);
  const int src = ei[((long)m * 2 + 0) * EE + e];
  const int dst = ei[((long)m * 2 + 1) * EE + e];
  const float4 a = *(const float4*)&al[((long)m * NN + dst) * HH];
  const float4 r = *(const float4*)&ar[((long)m * NN + src) * HH];
  float lg[HH] = {lrelu(a.x + r.x), lrelu(a.y + r.y),
                  lrelu(a.z + r.z), lrelu(a.w + r.w)};
  float* mb = &maxb[((long)m * NN + dst) * HH];
#pragma unroll
  for (int hd = 0; hd < HH; ++hd)
    (void)__hip_atomic_fetch_max(&mb[hd], lg[hd], __ATOMIC_RELAXED,
                                 __HIP_MEMORY_SCOPE_AGENT);
}

// ---------------------------------------------------------------------------
// 4) edge pass 2: segment sum of exp(logit - max)
// ---------------------------------------------------------------------------
__global__ void latte_edge_sum(const int* __restrict__ ei,
                               const float* __restrict__ al,
                               const float* __restrict__ ar,
                               const float* __restrict__ maxb,
                               float* __restrict__ sumb) {
  long idx = (long)blockIdx.x * blockDim.x + threadIdx.x;
  if (idx >= (long)MM * EE) return;
  const int m = (int)(idx / EE);
  const int e = (int)(idx % EE);
  const int src = ei[((long)m * 2 + 0) * EE + e];
  const int dst = ei[((long)m * 2 + 1) * EE + e];
  const float4 a = *(const float4*)&al[((long)m * NN + dst) * HH];
  const float4 r = *(const float4*)&ar[((long)m * NN + src) * HH];
  const float4 mx = *(const float4*)&maxb[((long)m * NN + dst) * HH];
  float ex[HH] = {__expf(lrelu(a.x + r.x) - mx.x), __expf(lrelu(a.y + r.y) - mx.y),
                  __expf(lrelu(a.z + r.z) - mx.z), __expf(lrelu(a.w + r.w) - mx.w)};
  float* sb = &sumb[((long)m * NN + dst) * HH];
#pragma unroll
  for (int hd = 0; hd < HH; ++hd) atomicAdd(&sb[hd], ex[hd]);
}

// ---------------------------------------------------------------------------
// 5) edge pass 3: rel[m,dst,:] += alpha * h[src,:], one wave per (m,e)
//    lane = channel; 4 coalesced f32 atomic-adds per lane (one per head)
// ---------------------------------------------------------------------------
__global__ void latte_edge_aggr(const int* __restrict__ ei,
                                const float* __restrict__ h,
                                const float* __restrict__ al,
                                const float* __restrict__ ar,
                                const float* __restrict__ maxb,
                                const float* __restrict__ sumb,
                                float* __restrict__ rel) {
  const int wave = threadIdx.x >> 5;
  const int lane = threadIdx.x & 31;
  const int e = blockIdx.x * 4 + wave;   // EE divisible by 4
  const int m = blockIdx.y;
  const int src = ei[((long)m * 2 + 0) * EE + e];
  const int dst = ei[((long)m * 2 + 1) * EE + e];
  const long db = ((long)m * NN + dst) * HH;
  const long sb = ((long)m * NN + src) * HH;
  float alpha[HH];
#pragma unroll
  for (int hd = 0; hd < HH; ++hd) {
    const float lg = lrelu(al[db + hd] + ar[sb + hd]);
    alpha[hd] = __expf(lg - maxb[db + hd]) / (sumb[db + hd] + 1e-16f);
  }
  float* rb = &rel[((long)m * NN + dst) * DD];
  const float* hs = &h[(long)src * DD];
#pragma unroll
  for (int hd = 0; hd < HH; ++hd)
    atomicAdd(&rb[hd * CC + lane], hs[hd * CC + lane] * alpha[hd]);
}

// ---------------------------------------------------------------------------
// 6) relation attention + combine; one block(128) per node, one wave per head
// ---------------------------------------------------------------------------
__global__ void latte_beta(const float* __restrict__ h,
                           const float* __restrict__ rel,
                           const float* __restrict__ ral,
                           const float* __restrict__ rar,
                           const float* __restrict__ rbias,
                           float* __restrict__ out) {
  const int n = blockIdx.x;
  const int t = threadIdx.x;             // 0..127 == hd*32+lane
  const float hv = h[(long)n * DD + t];
  float bl = hv * ral[t];
  bl = bl > 0.f ? bl : 0.f;

  float emb[MM + 1];
#pragma unroll
  for (int r = 0; r < MM; ++r) emb[r] = rel[((long)r * NN + n) * DD + t];
  emb[MM] = hv;

  float beta[MM + 1];
#pragma unroll
  for (int r = 0; r <= MM; ++r) {
    float br = emb[r] * rar[r * DD + t];
    br = br > 0.f ? br : 0.f;
    float p = bl * br;
#pragma unroll
    for (int off = 16; off > 0; off >>= 1) p += __shfl_xor(p, off, 32);
    beta[r] = p + rbias[r];              // butterfly: full sum in every lane
  }
  float mx = beta[0];
#pragma unroll
  for (int r = 1; r <= MM; ++r) mx = fmaxf(mx, beta[r]);
  float s = 0.f, acc = 0.f;
#pragma unroll
  for (int r = 0; r <= MM; ++r) {
    const float ev = __expf(beta[r] - mx);
    s += ev;
    acc += emb[r] * ev;
  }
  acc /= s;
  out[(long)n * DD + t] = acc > 0.f ? acc : 0.f;
}

// ---------------------------------------------------------------------------
extern "C" void kernel_launch(void* const* d_in, const int* in_sizes, int n_in,
                              void* d_out, int out_size, void* d_ws, size_t ws_size,
                              hipStream_t stream) {
  const float* feats = (const float*)d_in[0];
  const int*   ei    = (const int*)d_in[1];
  const float* W     = (const float*)d_in[2];
  const float* b     = (const float*)d_in[3];
  const float* attn  = (const float*)d_in[4];
  const float* ral   = (const float*)d_in[5];
  const float* rar   = (const float*)d_in[6];
  const float* rbias = (const float*)d_in[7];
  float* out = (float*)d_out;

  // workspace layout (floats)
  float* ws   = (float*)d_ws;
  float* h    = ws;                        // N*D        = 6,400,000
  float* al   = h    + (long)NN * DD;      // M*N*H      =   800,000
  float* ar   = al   + (long)MM * NN * HH;
  float* maxb = ar   + (long)MM * NN * HH;
  float* sumb = maxb + (long)MM * NN * HH;
  float* rel  = sumb + (long)MM * NN * HH; // M*N*D      = 25,600,000

  // 0) init accumulators (every call — deterministic)
  latte_init<<<4096, 256, 0, stream>>>(maxb, sumb, rel);

  // 1) projection GEMM via f32 WMMA; one wave per 16-row panel (TDM-staged A)
  latte_proj_wmma<<<NN / 16, 32, 0, stream>>>(feats, W, b, h);

  // 2) per-node attention scalars
  latte_scores<<<NN / 4, 128, 0, stream>>>(h, attn, al, ar);

  // 3-4) segment max / segment sum over edges
  const long mw = (long)MM * EE;
  latte_edge_max<<<(unsigned)((mw + 255) / 256), 256, 0, stream>>>(ei, al, ar, maxb);
  latte_edge_sum<<<(unsigned)((mw + 255) / 256), 256, 0, stream>>>(ei, al, ar, maxb, sumb);

  // 5) weighted scatter-add aggregation, one wave per (m, edge)
  latte_edge_aggr<<<dim3(EE / 4, MM), 128, 0, stream>>>(ei, h, al, ar, maxb, sumb, rel);

  // 6) relation-attention combine
  latte_beta<<<NN, DD, 0, stream>>>(h, rel, ral, rar, rbias, out);
}